// MaxMixturePrior_75333726372147
// MI455X (gfx1250) — compile-verified
//
#include <hip/hip_runtime.h>

typedef float v2f __attribute__((ext_vector_type(2)));
typedef float v8f __attribute__((ext_vector_type(8)));

#define DD    69      // pose dimension
#define MM    8       // num gaussians
#define KP    72      // K padded to multiple of 4
#define NPC   80      // padded columns per component (70 -> 80)
#define NTOT  (MM*NPC)
#define XS    76      // LDS x stride (floats): 76 mod 64 = 12 -> 16 distinct banks
#define BSTR  76      // LDS B stride (floats)
#define LOG2PI 1.83787706640934548356f

// ---------------------------------------------------------------------------
// Setup: build Qcat^T [NTOT][KP] (column-major in K) and -log(w) in workspace.
// Q_m = [[P, -P mu],[-(P mu)^T, mu^T P mu + 0.5*(logdet + D*log2pi)]]
// ---------------------------------------------------------------------------
__global__ __launch_bounds__(256) void gmm_setup_kernel(
    const float* __restrict__ means,      // [M,D]
    const float* __restrict__ prec,       // [M,D,D]
    const float* __restrict__ weights,    // [1,M]
    const float* __restrict__ cov_dets,   // [M]
    float* __restrict__ BmatT,            // [NTOT*KP]
    float* __restrict__ nlw)              // [M]
{
  __shared__ float pmu[MM * DD];
  __shared__ float qc[MM];
  const int tid = threadIdx.x;

  for (int idx = tid; idx < MM * DD; idx += 256) {
    int m = idx / DD, i = idx - m * DD;
    float s = 0.f;
    for (int j = 0; j < DD; ++j)
      s += prec[(m * DD + j) * DD + i] * means[m * DD + j];   // (P^T mu)_i == (P mu)_i (symmetric)
    pmu[idx] = s;
  }
  __syncthreads();
  if (tid < MM) {
    float s = 0.f;
    for (int i = 0; i < DD; ++i) s += means[tid * DD + i] * pmu[tid * DD + i];
    qc[tid]  = s + 0.5f * (cov_dets[tid] + (float)DD * LOG2PI);
    nlw[tid] = -logf(weights[tid]);
  }
  __syncthreads();
  for (int idx = tid; idx < NTOT * KP; idx += 256) {
    int n = idx / KP, k = idx - n * KP;
    int m = n / NPC, i = n - m * NPC;
    float v = 0.f;
    if (i < DD && k < DD)            v = prec[(m * DD + k) * DD + i];  // Q[k][i] = P[k][i]
    else if (i < DD && k == DD)      v = -pmu[m * DD + i];
    else if (i == DD && k < DD)      v = -pmu[m * DD + k];
    else if (i == DD && k == DD)     v = qc[m];
    BmatT[idx] = v;
  }
}

// ---------------------------------------------------------------------------
// Main kernel: 8 waves/block, 16 rows/wave. fp32 WMMA 16x16x4 GEMM
// Y = Xaug * Qcat, then per-row dot with Xaug, 16-lane reduce, min/argmin.
// ---------------------------------------------------------------------------
__global__ __launch_bounds__(256) void gmm_main_kernel(
    const float* __restrict__ pose,    // [B,D]
    const float* __restrict__ BmatT,   // [NTOT*KP] (K contiguous per column)
    const float* __restrict__ nlw,     // [M]
    float* __restrict__ out)           // [B]
{
  __shared__ float lds_x[8][16 * XS];     // per-wave x~ tile, 38912 B
  __shared__ float lds_B[NPC * BSTR];     // per-component Q slice, 24320 B

  const int tid  = threadIdx.x;
  const int lane = tid & 31;
  const int wv   = tid >> 5;
  const int l15  = lane & 15;
  const int hi   = lane >> 4;
  const int rowbase = (blockIdx.x * 8 + wv) * 16;

  // ---- stage pose tile (16 rows x 69) into LDS; col 69 = 1, cols 70/71 = 0
  float* xw = &lds_x[wv][0];
  {
    const float* src = pose + (size_t)rowbase * DD;
    for (int idx = lane; idx < 16 * DD; idx += 32) {
      int r = idx / DD;
      int c = idx - r * DD;
      xw[r * XS + c] = src[idx];        // coalesced global read
    }
    if (lane < 16) {
      xw[lane * XS + DD]     = 1.0f;
      xw[lane * XS + DD + 1] = 0.0f;
      xw[lane * XS + DD + 2] = 0.0f;
    }
  }

  // ---- A fragments in registers: a[j] = x~[row=l15][k], x~[row][k+1], k=4j+2*hi
  v2f a[KP / 4];
  #pragma unroll
  for (int j = 0; j < KP / 4; ++j) {
    int k = 4 * j + 2 * hi;
    a[j] = *(const v2f*)&xw[l15 * XS + k];   // ds_load_b64, conflict-free
  }

  float minv[8];
  int   mini[8];
  #pragma unroll
  for (int v = 0; v < 8; ++v) { minv[v] = 3.4e38f; mini[v] = 0; }

  for (int m = 0; m < MM; ++m) {
    __syncthreads();
    // stage component m's B slice (80 cols x 72 k) into LDS, restrided to 76
    {
      const v2f* src = (const v2f*)(BmatT + (size_t)m * NPC * KP);
      for (int idx = tid; idx < NPC * (KP / 2); idx += 256) {
        int n  = idx / (KP / 2);
        int kk = idx - n * (KP / 2);
        *(v2f*)&lds_B[n * BSTR + 2 * kk] = src[n * (KP / 2) + kk];
      }
    }
    __syncthreads();

    v8f dot = {0.f, 0.f, 0.f, 0.f, 0.f, 0.f, 0.f, 0.f};
    #pragma unroll
    for (int nt = 0; nt < 5; ++nt) {
      v8f acc = {0.f, 0.f, 0.f, 0.f, 0.f, 0.f, 0.f, 0.f};
      #pragma unroll
      for (int j = 0; j < KP / 4; ++j) {
        int k = 4 * j + 2 * hi;
        v2f b = *(const v2f*)&lds_B[(nt * 16 + l15) * BSTR + k]; // conflict-free b64
        acc = __builtin_amdgcn_wmma_f32_16x16x4_f32(
            false, a[j], false, b, (short)0, acc, false, false);
      }
      // phase 2: dot[v] += Y[row, i] * x~[row, i], i = nt*16 + l15 (column in lane)
      #pragma unroll
      for (int v = 0; v < 8; ++v) {
        int i  = nt * 16 + l15;
        int ic = (i < KP) ? i : 0;                 // keep address in-bounds
        float xv = xw[(v + 8 * hi) * XS + ic];
        xv = (i < KP) ? xv : 0.0f;                 // padded cols contribute 0
        dot[v] += acc[v] * xv;
      }
    }
    // reduce the 16 columns held across each 16-lane half
    #pragma unroll
    for (int v = 0; v < 8; ++v) {
      float s = dot[v];
      s += __shfl_xor(s, 1);
      s += __shfl_xor(s, 2);
      s += __shfl_xor(s, 4);
      s += __shfl_xor(s, 8);
      if (s < minv[v]) { minv[v] = s; mini[v] = m; }  // first-min wins (strict <)
    }
  }

  // lanes 0 / 16 hold rows [rowbase..+7] / [rowbase+8..+15] replicated
  if (l15 == 0) {
    int rb = rowbase + hi * 8;
    #pragma unroll
    for (int v = 0; v < 8; ++v)
      out[rb + v] = minv[v] + nlw[mini[v]];
  }
}

// ---------------------------------------------------------------------------
extern "C" void kernel_launch(void* const* d_in, const int* in_sizes, int n_in,
                              void* d_out, int out_size, void* d_ws, size_t ws_size,
                              hipStream_t stream) {
  const float* pose     = (const float*)d_in[0];   // [B,69]
  const float* means    = (const float*)d_in[1];   // [8,69]
  const float* prec     = (const float*)d_in[2];   // [8,69,69]
  const float* weights  = (const float*)d_in[3];   // [1,8]
  const float* cov_dets = (const float*)d_in[4];   // [8]
  float* out = (float*)d_out;                      // [B]

  float* BmatT = (float*)d_ws;                     // NTOT*KP floats = 184320 B
  float* nlw   = BmatT + NTOT * KP;                // 8 floats

  const int B = in_sizes[0] / DD;                  // 262144
  const int nblocks = B / (8 * 16);                // 2048

  gmm_setup_kernel<<<1, 256, 0, stream>>>(means, prec, weights, cov_dets, BmatT, nlw);
  gmm_main_kernel<<<nblocks, 256, 0, stream>>>(pose, BmatT, nlw, out);
}